// Attention_37675453120597
// MI455X (gfx1250) — compile-verified
//
#include <hip/hip_runtime.h>

#define HEADS 4
#define DHEAD 128
#define PIX 4096            // 64*64
#define CIN 256
#define SCALE 0.08838834764831845f   // 128^-0.5

typedef __attribute__((ext_vector_type(2))) float v2f;
typedef __attribute__((ext_vector_type(8))) float v8f;

// ---------------------------------------------------------------------------
// Kernel 1: qk[o,p] = sum_c W[o,c] * fmap[c,p]   (1x1 conv as GEMM)
// Tile orientation: M = pixels, N = output channels, so C-tile stores are
// contiguous along the head dim (coalesced 64B per half-wave).
// blockIdx.y selects Q (0) vs K (1) -> scalar-uniform destination, no
// per-lane EXEC-mask branching.
// ---------------------------------------------------------------------------
__global__ __launch_bounds__(256) void proj_qk(const float* __restrict__ fmap,
                                               const float* __restrict__ W,
                                               float* __restrict__ Qd,
                                               float* __restrict__ Kd) {
    const int wave = threadIdx.x >> 5;
    const int lane = threadIdx.x & 31;
    const int M    = lane & 15;
    const int half = lane >> 4;

    // 8192 tiles per half: 256 pixel-tiles x 32 channel-tiles; 8 waves/block.
    const int tile = blockIdx.x * 8 + wave;
    const int pT = tile >> 5;                 // 0..255
    const int nT = tile & 31;                 // 0..31
    const int m0 = pT * 16;                   // pixel base (rows / M)
    const int n0 = nT * 16;                   // channel base within half (cols / N)
    const int o0 = blockIdx.y * 512 + n0;     // absolute output channel base

    v8f c = {};
    for (int k = 0; k < CIN; k += 4) {
        const int kk = k + 2 * half;
        v2f a, b;
        a.x = fmap[kk * PIX + m0 + M];        // A[M][k] = fmap[c][pixel]
        a.y = fmap[(kk + 1) * PIX + m0 + M];
        b.x = W[(o0 + M) * CIN + kk];         // B[k][N] = W[o][c] (contig -> b64)
        b.y = W[(o0 + M) * CIN + kk + 1];
        c = __builtin_amdgcn_wmma_f32_16x16x4_f32(false, a, false, b,
                                                  (short)0, c, false, false);
    }

    const int h  = n0 >> 7;                   // head (uniform per tile)
    const int d0 = n0 & 127;                  // dim base (uniform per tile)
    float* __restrict__ dst = blockIdx.y ? Kd : Qd;
    const float scl = blockIdx.y ? 1.0f : SCALE;

    // C element r = (pixel m0 + r + 8*half, channel o0 + M)
    for (int r = 0; r < 8; ++r) {
        const int p = m0 + r + 8 * half;
        dst[((h * PIX) + p) * DHEAD + d0 + M] = c[r] * scl;
    }
}

// ---------------------------------------------------------------------------
// Kernel 2: fused  S = Q K^T  (16 rows x 4096 cols per workgroup) + softmax.
// Score strip lives in LDS (16*4096*4B = 256 KB of the 320 KB WGP LDS).
// Q fragment is preloaded into 64 VGPRs per lane; inner WMMA loop only
// streams K via contiguous global_load_b64 (K stays L2-resident, 8 MB).
// Output written exactly once, non-temporally (write-once 256 MB).
// ---------------------------------------------------------------------------
__global__ __launch_bounds__(256) void attn_softmax(const float* __restrict__ Qd,
                                                    const float* __restrict__ Kd,
                                                    float* __restrict__ out) {
    __shared__ float sim[16 * PIX];     // 256 KB score strip
    __shared__ float qs[16 * DHEAD];    // 8 KB Q tile
    __shared__ float red[256];
    __shared__ float rstat[16];

    const int h  = blockIdx.y;          // head 0..3
    const int m0 = blockIdx.x * 16;     // row tile base
    const int wave = threadIdx.x >> 5;
    const int lane = threadIdx.x & 31;
    const int M    = lane & 15;
    const int half = lane >> 4;

    // Stage Q tile (16 rows x 128 dims), coalesced.
    for (int i = threadIdx.x; i < 16 * DHEAD; i += 256) {
        const int r = i >> 7, d = i & 127;
        qs[i] = Qd[((h * PIX) + (m0 + r)) * DHEAD + d];
    }
    __syncthreads();

    // Preload this lane's A-fragment: Q[M][2*half + {0,1} + 4*kv], kv=0..31.
    v2f areg[32];
    for (int kv = 0; kv < 32; ++kv) {
        const int kk = kv * 4 + 2 * half;
        areg[kv].x = qs[M * DHEAD + kk];
        areg[kv].y = qs[M * DHEAD + kk + 1];
    }

    // Compute score strip: 256 column tiles round-robined over 8 waves.
    for (int t = wave; t < PIX / 16; t += 8) {
        const int n0 = t * 16;
        const float* __restrict__ krow = &Kd[((h * PIX) + (n0 + M)) * DHEAD];
        v8f c = {};
        for (int kv = 0; kv < 32; ++kv) {
            const int kk = kv * 4 + 2 * half;
            v2f b;
            b.x = krow[kk];             // B = K^T: b[k][n] = K[n][k] (contig -> b64)
            b.y = krow[kk + 1];
            c = __builtin_amdgcn_wmma_f32_16x16x4_f32(false, areg[kv], false, b,
                                                      (short)0, c, false, false);
        }
        for (int r = 0; r < 8; ++r)
            sim[(r + 8 * half) * PIX + n0 + M] = c[r];
    }
    __syncthreads();

    // Softmax over each of the 16 rows; 16 threads per row.
    const int row = threadIdx.x >> 4;
    const int cid = threadIdx.x & 15;

    float m = -1e30f;
    for (int ccol = cid; ccol < PIX; ccol += 16)
        m = fmaxf(m, sim[row * PIX + ccol]);
    red[threadIdx.x] = m;
    __syncthreads();
    for (int s = 8; s > 0; s >>= 1) {
        if (cid < s)
            red[threadIdx.x] = fmaxf(red[threadIdx.x], red[threadIdx.x + s]);
        __syncthreads();
    }
    const float rowmax = red[row * 16];
    __syncthreads();

    float ssum = 0.0f;
    for (int ccol = cid; ccol < PIX; ccol += 16) {
        const float e = __expf(sim[row * PIX + ccol] - rowmax);
        sim[row * PIX + ccol] = e;
        ssum += e;
    }
    red[threadIdx.x] = ssum;
    __syncthreads();
    for (int s = 8; s > 0; s >>= 1) {
        if (cid < s)
            red[threadIdx.x] += red[threadIdx.x + s];
        __syncthreads();
    }
    if (cid == 0)
        rstat[row] = 1.0f / red[row * 16];
    __syncthreads();

    // Normalized, coalesced, non-temporal store (write-once stream).
    const size_t base = ((size_t)h * PIX + m0) * PIX;
    for (int idx = threadIdx.x; idx < 16 * PIX; idx += 256) {
        const int r = idx >> 12;                  // idx / 4096
        __builtin_nontemporal_store(sim[idx] * rstat[r], &out[base + idx]);
    }
}

extern "C" void kernel_launch(void* const* d_in, const int* in_sizes, int n_in,
                              void* d_out, int out_size, void* d_ws, size_t ws_size,
                              hipStream_t stream) {
    const float* fmap = (const float*)d_in[0];   // [1,256,64,64]
    const float* W    = (const float*)d_in[1];   // [1024,256]
    float* out = (float*)d_out;                  // [1,4,4096,4096]

    float* Qd = (float*)d_ws;                     // [4][4096][128]  (8 MB)
    float* Kd = Qd + (size_t)HEADS * PIX * DHEAD; // [4][4096][128]  (8 MB)

    // 8192 tiles per half (Q / K), 8 tiles (waves) per block, y = Q|K.
    proj_qk<<<dim3(1024, 2), 256, 0, stream>>>(fmap, W, Qd, Kd);

    // 256 row-tiles x 4 heads.
    attn_softmax<<<dim3(256, 4), 256, 0, stream>>>(Qd, Kd, out);
}